// Triplane_22162031247387
// MI455X (gfx1250) — compile-verified
//
#include <hip/hip_runtime.h>

// Triplane sampling (EG3D-style) for MI455X / gfx1250.
//
// Bottleneck analysis: output stream = 307 MB NT stores (HBM floor ~18us at
// 23.3 TB/s); plane set = 100.7 MB -> fully L2-resident (192 MB). The real
// limiter is divergent-gather request throughput (lane==channel => 32 distinct
// lines per gather instruction). So: minimize gather instruction/request count.
// x0/x1 corners are row-adjacent -> one b64 load per row instead of two b32,
// halving gather requests. WMMA is inapplicable: the bilinear combine is a
// batched (1x4)*(4x32) GEMM whose B operand differs per sample, so no wave
// tile shares operands.

namespace {
constexpr int KS = 4;
constexpr int CCH = 32;            // channels == lanes (wave32)
constexpr int RESO = 512;
constexpr int HW = RESO * RESO;    // floats per channel plane
constexpr int PLANE_STRIDE = CCH * HW;

struct __attribute__((aligned(4))) FPair { float a, b; };  // 4B-aligned 8B load
}

__global__ __launch_bounds__(256) void triplane_sample_kernel(
    const float* __restrict__ coords,   // (N, K, 3)
    const float* __restrict__ maxc,     // (3,)
    const float* __restrict__ minc,     // (3,)
    const float* __restrict__ planes,   // (3, C, 512, 512)
    float* __restrict__ out,            // (pair, p, c) flat
    int total_pairs)                    // N*K
{
  const int lane   = threadIdx.x & 31;                       // channel
  const int gwave  = (blockIdx.x * blockDim.x + threadIdx.x) >> 5;
  const int nwaves = (gridDim.x * blockDim.x) >> 5;

  // Per-plane 1/sqrt(mag_sq); mag_sq = min(bbx, (0.5*RADII)^2 = 0.25).
  const float mx0 = maxc[0], mx1 = maxc[1], mx2 = maxc[2];
  const float mn0 = minc[0], mn1 = minc[1], mn2 = minc[2];
  float inv[3];
  {
    const float bb0 = fminf(mx0 * mx0 + mx1 * mx1, mn0 * mn0 + mn1 * mn1);
    const float bb1 = fminf(mx0 * mx0 + mx2 * mx2, mn0 * mn0 + mn2 * mn2);
    const float bb2 = fminf(mx2 * mx2 + mx0 * mx0, mn2 * mn2 + mn0 * mn0);
    inv[0] = 1.0f / sqrtf(fminf(bb0, 0.25f));
    inv[1] = 1.0f / sqrtf(fminf(bb1, 0.25f));
    inv[2] = 1.0f / sqrtf(fminf(bb2, 0.25f));
  }

  for (int pair = gwave; pair < total_pairs; pair += nwaves) {
    const float* cp = coords + (size_t)pair * 3;
    const float c0 = 2.0f * __builtin_nontemporal_load(cp + 0);  // 2/box_warp
    const float c1 = 2.0f * __builtin_nontemporal_load(cp + 1);
    const float c2 = 2.0f * __builtin_nontemporal_load(cp + 2);
    if (pair + nwaves < total_pairs)
      __builtin_prefetch(coords + (size_t)(pair + nwaves) * 3, 0, 0);

    // Decomposed projections: p0:(y,z) p1:(x,z) p2:(x,y)
    const float pxa[3] = {c1, c0, c0};
    const float pya[3] = {c2, c2, c1};
    float* op = out + (size_t)pair * (3 * CCH);

#pragma unroll
    for (int p = 0; p < 3; ++p) {
      // nc = (proj/sqrt(mag_sq))*2 - 1, then contract(nc*6)*0.5
      float gx = fmaf(pxa[p] * inv[p], 2.0f, -1.0f) * 6.0f;
      float gy = fmaf(pya[p] * inv[p], 2.0f, -1.0f) * 6.0f;
      const float m = fmaxf(gx * gx + gy * gy, 1.1920929e-7f);
      if (m > 1.0f) {                       // wave-uniform branch
        const float s = (2.0f * sqrtf(m) - 1.0f) / m;
        gx *= s;
        gy *= s;
      }
      gx *= 0.5f;
      gy *= 0.5f;

      // grid_sample bilinear, zeros padding, align_corners=False
      const float ix = fmaf(gx + 1.0f, (float)RESO, -1.0f) * 0.5f;
      const float iy = fmaf(gy + 1.0f, (float)RESO, -1.0f) * 0.5f;
      const float x0f = floorf(ix), y0f = floorf(iy);
      const float wx1 = ix - x0f, wy1 = iy - y0f;
      const float wx0 = 1.0f - wx1, wy0 = 1.0f - wy1;
      const int x0 = (int)x0f, y0 = (int)y0f;
      const int x1 = x0 + 1, y1 = y0 + 1;

      const float vy0 = (y0 >= 0 && y0 < RESO) ? 1.0f : 0.0f;
      const float vy1 = (y1 >= 0 && y1 < RESO) ? 1.0f : 0.0f;
      const int yc0 = min(max(y0, 0), RESO - 1);
      const int yc1 = min(max(y1, 0), RESO - 1);

      const float* fb = planes + (size_t)p * PLANE_STRIDE + (size_t)lane * HW;

      float f00, f10, f01, f11;
      if (__builtin_expect(x0 >= 0 && x1 < RESO, 1)) {
        // Fast path (~all samples): x corners are row-adjacent -> one 8B
        // gather per row. Halves divergent L2 request count vs 2x b32.
        const FPair r0 = *(const FPair*)(fb + yc0 * RESO + x0);
        const FPair r1 = *(const FPair*)(fb + yc1 * RESO + x0);
        f00 = r0.a * vy0;
        f10 = r0.b * vy0;
        f01 = r1.a * vy1;
        f11 = r1.b * vy1;
      } else {
        // Edge path: clamped scalar gathers with zero-padding masks.
        const float vx0 = (x0 >= 0 && x0 < RESO) ? 1.0f : 0.0f;
        const float vx1 = (x1 >= 0 && x1 < RESO) ? 1.0f : 0.0f;
        const int xc0 = min(max(x0, 0), RESO - 1);
        const int xc1 = min(max(x1, 0), RESO - 1);
        f00 = fb[yc0 * RESO + xc0] * (vx0 * vy0);
        f10 = fb[yc0 * RESO + xc1] * (vx1 * vy0);
        f01 = fb[yc1 * RESO + xc0] * (vx0 * vy1);
        f11 = fb[yc1 * RESO + xc1] * (vx1 * vy1);
      }

      const float r = f00 * (wx0 * wy0) + f10 * (wx1 * wy0) +
                      f01 * (wx0 * wy1) + f11 * (wx1 * wy1);

      // NT store: don't let the 307 MB output stream evict the plane set.
      __builtin_nontemporal_store(r, op + p * CCH + lane);
    }
  }
}

extern "C" void kernel_launch(void* const* d_in, const int* in_sizes, int n_in,
                              void* d_out, int out_size, void* d_ws, size_t ws_size,
                              hipStream_t stream) {
  const float* coords = (const float*)d_in[0];   // (N,K,3) f32
  const float* maxc   = (const float*)d_in[1];   // (3,)
  const float* minc   = (const float*)d_in[2];   // (3,)
  const float* planes = (const float*)d_in[3];   // (3,C,512,512) f32
  float* out = (float*)d_out;

  const int total_pairs = in_sizes[0] / 3;       // N*K = 800000
  const int threads = 256;                       // 8 wave32 per block
  int blocks = (total_pairs + 31) / 32;          // ~4 pairs per wave
  if (blocks < 1) blocks = 1;

  triplane_sample_kernel<<<blocks, threads, 0, stream>>>(
      coords, maxc, minc, planes, out, total_pairs);
}